// SoftAttention_25898652795668
// MI455X (gfx1250) — compile-verified
//
#include <hip/hip_runtime.h>

typedef __attribute__((ext_vector_type(2))) float v2f;
typedef __attribute__((ext_vector_type(8))) float v8f;

// Math identity: softmax(x, -1).sum(-1) == 1, so score == ones and each
// branch's context is sum_p emb[b,p,:], broadcast over S. The surviving GEMM
// (ones[SxP] @ emb[Pxd]) is computed with V_WMMA_F32_16X16X4_F32:
// A = ones(16x4), B = 4x16 tiles of emb, K accumulated over P=64 (16 wmma).
// Since A is all-ones the result is invariant to fragment K/M permutation and
// every D row equals the column sums.
//
// One wave per block; block = (branch, batch, 16-column tile):
// grid = 2 * 4 * (512/16) = 256 blocks of 32 threads.
__global__ __launch_bounds__(32)
void softattn_collapsed_wmma(const float* __restrict__ syn,
                             const float* __restrict__ pos,
                             float* __restrict__ out) {
    const int bid  = blockIdx.x;        // 0..255
    const int br   = bid >> 7;          // 0 = syn branch, 1 = pos branch
    const int b    = (bid >> 5) & 3;    // batch 0..3
    const int nt   = bid & 31;          // 16-column tile 0..31
    const int lane = threadIdx.x;       // 0..31

    const int col  = nt * 16 + (lane & 15);  // d-column (duplicated across halves)
    const int krow = (lane >> 4) * 2;        // K rows 0,1 / 2,3 per half-wave

    const float* __restrict__ emb = (br == 0 ? syn : pos) + (size_t)b * 64 * 512;

    v2f a; a.x = 1.0f; a.y = 1.0f;           // A = ones(16x4)
    v8f c = {};

    #pragma unroll
    for (int kt = 0; kt < 16; ++kt) {
        const int p0 = kt * 4 + krow;
        v2f bb;
        bb.x = emb[(size_t)p0 * 512 + col];
        bb.y = emb[(size_t)(p0 + 1) * 512 + col];
        c = __builtin_amdgcn_wmma_f32_16x16x4_f32(
                /*neg_a=*/false, a, /*neg_b=*/false, bb,
                /*c_mod=*/(short)0, c, /*reuse_a=*/false, /*reuse_b=*/false);
    }

    // c[0] = sum_{p=0..63} emb[b, p, col], identical in both wave halves.
    const float colsum = c[0];

    // Repack for wide stores: lane L takes 4-column group g = (L&3) and s-row
    // subindex (L>>2) in 0..7. Gather the group's 4 column sums via shuffles
    // (sources are lanes 0..15, which own cols 0..15 of this tile).
    const int g4 = (lane & 3) * 4;
    float4 v;
    v.x = __shfl(colsum, g4 + 0, 32);
    v.y = __shfl(colsum, g4 + 1, 32);
    v.z = __shfl(colsum, g4 + 2, 32);
    v.w = __shfl(colsum, g4 + 3, 32);

    // Each wave-iteration stores 8 full s-rows of this 16-col tile (128 floats,
    // contiguous 64B per lane-quad). 256 s-rows -> 32 fully-unrolled B128
    // stores with immediate offsets (i*32768 B < 2^23, fits IOFFSET).
    float* __restrict__ outp = out + (size_t)b * 256 * 1024
                                   + (size_t)(lane >> 2) * 1024
                                   + (size_t)br * 512 + nt * 16 + g4;
    #pragma unroll
    for (int i = 0; i < 32; ++i) {
        *(float4*)(outp + (size_t)i * 8 * 1024) = v;
    }
}

extern "C" void kernel_launch(void* const* d_in, const int* in_sizes, int n_in,
                              void* d_out, int out_size, void* d_ws, size_t ws_size,
                              hipStream_t stream) {
    // setup_inputs order: [0]=syn_emb [4,64,512], [1]=pos_emb [4,64,512],
    // [2]=dec, [3..10]=weights/biases (analytically irrelevant: softmax sums to 1).
    const float* syn = (const float*)d_in[0];
    const float* pos = (const float*)d_in[1];
    float* out = (float*)d_out;   // [4, 256, 1024] f32

    softattn_collapsed_wmma<<<256, 32, 0, stream>>>(syn, pos, out);
}